// AnalyticMaxwellRNNCell_13958643712566
// MI455X (gfx1250) — compile-verified
//
#include <hip/hip_runtime.h>
#include <cstdint>

// ---------------------------------------------------------------------------
// Maxwell viscoelastic recurrence (linear 1st-order scan), B=2048 rows, T=8192.
//   gamma' = a*gamma + b,  a = 1-2*dt, b = 2*dt*eps,  sig = 2.5*eps - 2*gamma
// Memory-bound: 192 MiB total traffic -> ~8.6 us floor @ 23.3 TB/s.
// One wave32 per row; tiles of 256 timesteps streamed into LDS by the
// Tensor Data Mover (double-buffered), wave-parallel affine scan via shuffles.
// ---------------------------------------------------------------------------

typedef unsigned int u32x4 __attribute__((ext_vector_type(4)));
typedef int          i32x8 __attribute__((ext_vector_type(8)));
typedef int          i32x4 __attribute__((ext_vector_type(4)));

#define LANES 32
#define WAVES 8
#define TILE  256            // timesteps per wave-tile
#define EPL   (TILE / LANES) // 8 elements per lane

// Issue a TDM load of a 1-row tile of `nelem` f32 from global -> LDS.
// D# layout per CDNA5 ISA 8.3/8.4: group0 = {count=1 | lds_addr | global_addr
// | type=2}, group1 = {data_size=4B, tensor_dim0=nelem, tensor_dim1=1,
// tile_dim0=nelem, tile_dim1=1, dim0_stride=nelem}. Groups 2/3 zero (2D).
__device__ __forceinline__ void tdm_load_1d(uint32_t lds_byte, const float* gsrc,
                                            int nelem) {
  uint64_t ga = (uint64_t)(uintptr_t)gsrc;
  u32x4 g0;
  g0.x = 1u;                                   // count=1, user descriptor
  g0.y = lds_byte;                             // LDS byte address
  g0.z = (uint32_t)ga;                         // global_addr[31:0]
  g0.w = (uint32_t)(ga >> 32) | 0x80000000u;   // global_addr[56:32] | type=2
  i32x8 g1;
  g1[0] = 0x00020000;                          // data_size=2 (4 bytes), no mask
  g1[1] = (nelem & 0xFFFF) << 16;              // tensor_dim0[15:0] @ bits 63:48
  g1[2] = ((nelem >> 16) & 0xFFFF) | (1 << 16);// tensor_dim0[31:16] | tensor_dim1=1
  g1[3] = (nelem & 0xFFFF) << 16;              // tile_dim0 @ bits 127:112
  g1[4] = 1;                                   // tile_dim1=1, tile_dim2=0
  g1[5] = nelem;                               // tensor_dim0_stride[31:0]
  g1[6] = 0;
  g1[7] = 0;
  i32x4 z4 = {0, 0, 0, 0};
  i32x8 z8 = {0, 0, 0, 0, 0, 0, 0, 0};
  __builtin_amdgcn_tensor_load_to_lds(g0, g1, z4, z4, z8, 0);
}

__global__ void __launch_bounds__(WAVES * LANES)
maxwell_scan_kernel(const float* __restrict__ eps, const float* __restrict__ dts,
                    float* __restrict__ out, int T, int B) {
  __shared__ float lds[WAVES][2][2][TILE];  // [wave][buf][eps/dts][TILE]

  const int lane = threadIdx.x & (LANES - 1);
  const int wave = threadIdx.x >> 5;
  const int row  = blockIdx.x * WAVES + wave;
  if (row >= B) return;  // wave-uniform; no TDM issued by inactive waves

  const size_t base = (size_t)row * (size_t)T;
  const float* erow = eps + base;
  const float* drow = dts + base;
  float*       orow = out + base;
  const int nTiles = T / TILE;

  uint32_t lds_e[2], lds_d[2];
  lds_e[0] = (uint32_t)(uintptr_t)&lds[wave][0][0][0];
  lds_d[0] = (uint32_t)(uintptr_t)&lds[wave][0][1][0];
  lds_e[1] = (uint32_t)(uintptr_t)&lds[wave][1][0][0];
  lds_d[1] = (uint32_t)(uintptr_t)&lds[wave][1][1][0];

  // Prologue: prefetch tiles 0 and 1 (2 TDM ops each).
  tdm_load_1d(lds_e[0], erow, TILE);
  tdm_load_1d(lds_d[0], drow, TILE);
  if (nTiles > 1) {
    tdm_load_1d(lds_e[1], erow + TILE, TILE);
    tdm_load_1d(lds_d[1], drow + TILE, TILE);
  }

  float g_carry = 0.0f;  // gamma entering the current tile (uniform per wave)

  for (int t = 0; t < nTiles; ++t) {
    const int buf = t & 1;

    // TDM ops from one wave complete in order: waiting TENSORcnt<=2 releases
    // tile t while tile t+1's two loads remain outstanding.
    if (t + 1 < nTiles) __builtin_amdgcn_s_wait_tensorcnt(2);
    else                __builtin_amdgcn_s_wait_tensorcnt(0);

    // Load this lane's 8 contiguous timesteps from LDS (2x b128 reads each).
    float e[EPL], d[EPL];
    const float* le = &lds[wave][buf][0][lane * EPL];
    const float* ld = &lds[wave][buf][1][lane * EPL];
#pragma unroll
    for (int j = 0; j < EPL; j += 4) {
      float4 ev = *(const float4*)(le + j);
      float4 dv = *(const float4*)(ld + j);
      e[j] = ev.x; e[j + 1] = ev.y; e[j + 2] = ev.z; e[j + 3] = ev.w;
      d[j] = dv.x; d[j + 1] = dv.y; d[j + 2] = dv.z; d[j + 3] = dv.w;
    }

    // Per-lane composition of 8 affine steps: g -> A*g + Bv.
    float A = 1.0f, Bv = 0.0f;
#pragma unroll
    for (int j = 0; j < EPL; ++j) {
      float a = 1.0f - 2.0f * d[j];
      float b = 2.0f * d[j] * e[j];
      Bv = a * Bv + b;
      A  = a * A;
    }

    // Inclusive wave32 scan under affine composition (Hillis-Steele, 5 steps).
#pragma unroll
    for (int off = 1; off < LANES; off <<= 1) {
      float Ap = __shfl_up(A, off, LANES);
      float Bp = __shfl_up(Bv, off, LANES);
      if (lane >= off) { Bv = A * Bp + Bv; A = A * Ap; }
    }

    // Exclusive prefix for this lane -> gamma at this lane's first timestep.
    float Ae = __shfl_up(A, 1, LANES);
    float Be = __shfl_up(Bv, 1, LANES);
    if (lane == 0) { Ae = 1.0f; Be = 0.0f; }
    float g = Ae * g_carry + Be;

    // Carry for the next tile from lane 31's inclusive result.
    float A31 = __shfl(A, LANES - 1, LANES);
    float B31 = __shfl(Bv, LANES - 1, LANES);
    g_carry = A31 * g_carry + B31;

    // Replay the 8 local steps, emitting sig = 2.5*eps - 2*gamma.
    float s[EPL];
#pragma unroll
    for (int j = 0; j < EPL; ++j) {
      s[j] = 2.5f * e[j] - 2.0f * g;
      g = g + 2.0f * d[j] * (e[j] - g);
    }

    // Coalesced b128 stores (lane l covers bytes [l*32, l*32+32) of the tile).
    float* op = orow + (size_t)t * TILE + lane * EPL;
#pragma unroll
    for (int j = 0; j < EPL; j += 4) {
      float4 sv = {s[j], s[j + 1], s[j + 2], s[j + 3]};
      *(float4*)(op + j) = sv;
    }

    // Prefetch tile t+2 into the buffer we just finished consuming.
    if (t + 2 < nTiles) {
      tdm_load_1d(lds_e[buf], erow + (size_t)(t + 2) * TILE, TILE);
      tdm_load_1d(lds_d[buf], drow + (size_t)(t + 2) * TILE, TILE);
    }
  }
}

extern "C" void kernel_launch(void* const* d_in, const int* in_sizes, int n_in,
                              void* d_out, int out_size, void* d_ws, size_t ws_size,
                              hipStream_t stream) {
  const float* eps = (const float*)d_in[0];
  const float* dts = (const float*)d_in[1];
  float* out = (float*)d_out;

  const int T = 8192;                 // reference fixes T
  const int B = in_sizes[0] / T;      // = 2048

  dim3 block(WAVES * LANES);          // 256 threads = 8 wave32
  dim3 grid((B + WAVES - 1) / WAVES); // one row per wave
  maxwell_scan_kernel<<<grid, block, 0, stream>>>(eps, dts, out, T, B);
}